// AttFusion_46634754900065
// MI455X (gfx1250) — compile-verified
//
#include <hip/hip_runtime.h>
#include <cstdint>

// Problem constants (fixed in the reference).
constexpr int C = 256;
constexpr int N = 12288;              // W*H = 64*192
constexpr int NTILE = 32;             // n positions per block (one 128B line per (l,c) row)
constexpr int NWAVES = 16;            // waves per block
constexpr int THREADS = NWAVES * 32;  // 512
constexpr int CPT = C / NWAVES;       // 16 channels per thread
constexpr int NCHUNK = CPT / 2;       // 8 chunks of 2 channels

__device__ __forceinline__ unsigned lds_addr32(const void* p) {
  // Generic shared pointer: low 32 bits are the LDS byte offset (aperture is in the high half).
  return (unsigned)(uintptr_t)p;
}

__device__ __forceinline__ void async_load_b64(unsigned lds_byte, unsigned voff,
                                               const void* base) {
  // GVS mode: mem = SADDR + VADDR[31:0]; dest LDS byte address in the first VGPR.
  asm volatile("global_load_async_to_lds_b64 %0, %1, %2"
               :: "v"(lds_byte), "v"(voff), "s"(base) : "memory");
}

template <int NIF>
__device__ __forceinline__ void wait_async() {
  asm volatile("s_wait_asynccnt %0" :: "n"(NIF) : "memory");
}

__device__ __forceinline__ void wait_ds0() {
  asm volatile("s_wait_dscnt 0" ::: "memory");
}

__host__ __device__ constexpr int PI(int l, int m, int LL) {
  // index of pair (l,m), l<=m, in row-major upper-triangle enumeration
  return l * LL - (l * (l - 1)) / 2 + (m - l);
}

template <int L>
__launch_bounds__(THREADS)
__global__ void group_attn(const float* __restrict__ x, int rec0, int rec1,
                           float* __restrict__ out, int g0, int g1,
                           float* __restrict__ attn1) {
  constexpr bool TRI = (L == 2);                 // full gram only needed for the attn-map group
  constexpr int NP = TRI ? (L * (L + 1)) / 2 : L;
  constexpr int CHUNK_FLOATS = L * 2 * NTILE;    // per-wave chunk: L rows x 2 channels x 32 n

  __shared__ __align__(16) float stage[NWAVES * 2 * CHUNK_FLOATS];  // wave-private dbl buffers
  __shared__ float red[NP][NWAVES][NTILE];
  __shared__ float wls[L][NTILE];

  const int tid = (int)threadIdx.x;
  const int tx = tid & 31;     // n within tile == lane id
  const int ty = tid >> 5;     // wave id == channel slice
  const int n0 = (int)blockIdx.x * NTILE;
  const int gy = (int)blockIdx.y;

  const float* xg = x + (size_t)(gy ? rec1 : rec0) * C * N;  // group base [L][C][N]
  float* outg = out + (size_t)(gy ? g1 : g0) * C * N;        // [C][N]
  float* attn = (TRI && gy) ? attn1 : nullptr;               // only group 7

  const int c0 = ty * CPT;

  // --- async DMA staging: global -> LDS (wave-private, double buffered) -> VGPR ---
  const unsigned waveB = lds_addr32(stage) + (unsigned)(ty * 2 * CHUNK_FLOATS) * 4u;
  const unsigned chunkB = (unsigned)CHUNK_FLOATS * 4u;
  const int ccl = tx >> 4;             // DMA lane: which of the 2 channels in the chunk
  const int nel = (tx & 15) << 1;      // DMA lane: first of its 2 n elements
  const unsigned nByte = (unsigned)(n0 + nel) * 4u;

  float v[L][CPT];

  // issue chunk 0  (L b64 instructions: each covers 2 rows of 128B)
#pragma unroll
  for (int l = 0; l < L; ++l) {
    const unsigned g = ((unsigned)((l * C + c0 + ccl) * N)) * 4u + nByte;
    const unsigned d = waveB + (unsigned)(((l * 2 + ccl) * NTILE + nel) * 4);
    async_load_b64(d, g, xg);
  }

#pragma unroll
  for (int k = 0; k < NCHUNK; ++k) {
    if (k + 1 < NCHUNK) {
      wait_ds0();  // make sure prior LDS reads of the buffer we are about to overwrite retired
      const unsigned pB = waveB + (unsigned)((k + 1) & 1) * chunkB;
#pragma unroll
      for (int l = 0; l < L; ++l) {
        const unsigned g = ((unsigned)((l * C + c0 + 2 * (k + 1) + ccl) * N)) * 4u + nByte;
        const unsigned d = pB + (unsigned)(((l * 2 + ccl) * NTILE + nel) * 4);
        async_load_b64(d, g, xg);
      }
      wait_async<L>();   // <=L in flight -> chunk k (issued first, completes in order) is done
    } else {
      wait_async<0>();
    }
    const float* sb = &stage[(ty * 2 + (k & 1)) * CHUNK_FLOATS];
#pragma unroll
    for (int l = 0; l < L; ++l) {
      v[l][2 * k + 0] = sb[(l * 2 + 0) * NTILE + tx];
      v[l][2 * k + 1] = sb[(l * 2 + 1) * NTILE + tx];
    }
  }

  // --- partial channel dot products ---
  float pd[NP];
#pragma unroll
  for (int p = 0; p < NP; ++p) pd[p] = 0.0f;
#pragma unroll
  for (int k = 0; k < CPT; ++k) {
    if (TRI) {
#pragma unroll
      for (int l = 0; l < L; ++l)
#pragma unroll
        for (int m = l; m < L; ++m)
          pd[PI(l, m, L)] = fmaf(v[l][k], v[m][k], pd[PI(l, m, L)]);
    } else {
#pragma unroll
      for (int m = 0; m < L; ++m)
        pd[m] = fmaf(v[0][k], v[m][k], pd[m]);
    }
  }
#pragma unroll
  for (int p = 0; p < NP; ++p) red[p][ty][tx] = pd[p];
  __syncthreads();

  // --- wave 0: reduce across channel slices, softmax, broadcast weights ---
  if (ty == 0) {
    float dtot[NP];
#pragma unroll
    for (int p = 0; p < NP; ++p) {
      float s = 0.0f;
#pragma unroll
      for (int w = 0; w < NWAVES; ++w) s += red[p][w][tx];
      dtot[p] = s;
    }
    // row-0 softmax (scale 1/sqrt(256) = 0.0625), pairs (0,m) are indices 0..L-1
    float s0[L], smax = -3.0e38f;
#pragma unroll
    for (int m = 0; m < L; ++m) { s0[m] = dtot[m] * 0.0625f; smax = fmaxf(smax, s0[m]); }
    float e0[L], Z = 0.0f;
#pragma unroll
    for (int m = 0; m < L; ++m) { e0[m] = __expf(s0[m] - smax); Z += e0[m]; }
    const float iZ = 1.0f / Z;
#pragma unroll
    for (int m = 0; m < L; ++m) wls[m][tx] = e0[m] * iZ;

    if (TRI && attn) {  // full LxL attention map for the last group (L==2)
#pragma unroll
      for (int l = 0; l < L; ++l) {
        float sr[L], rmax = -3.0e38f;
#pragma unroll
        for (int m = 0; m < L; ++m) {
          const int a = l < m ? l : m, b = l < m ? m : l;
          sr[m] = dtot[PI(a, b, L)] * 0.0625f;
          rmax = fmaxf(rmax, sr[m]);
        }
        float er[L], Zr = 0.0f;
#pragma unroll
        for (int m = 0; m < L; ++m) { er[m] = __expf(sr[m] - rmax); Zr += er[m]; }
        const float iZr = 1.0f / Zr;
#pragma unroll
        for (int m = 0; m < L; ++m)
          attn[(size_t)(l * L + m) * N + n0 + tx] = er[m] * iZr;
      }
    }
  }
  __syncthreads();

  // --- weighted combination from registers, coalesced stores ---
  float w[L];
#pragma unroll
  for (int m = 0; m < L; ++m) w[m] = wls[m][tx];
#pragma unroll
  for (int k = 0; k < CPT; ++k) {
    float acc = 0.0f;
#pragma unroll
    for (int m = 0; m < L; ++m) acc = fmaf(w[m], v[m][k], acc);
    outg[(size_t)(c0 + k) * N + n0 + tx] = acc;
  }
}

extern "C" void kernel_launch(void* const* d_in, const int* in_sizes, int n_in,
                              void* d_out, int out_size, void* d_ws, size_t ws_size,
                              hipStream_t stream) {
  (void)in_sizes; (void)n_in; (void)out_size; (void)d_ws; (void)ws_size;
  const float* x = (const float*)d_in[0];
  float* out = (float*)d_out;
  float* attn = out + (size_t)8 * C * N;   // attn maps of last group appended

  // RECORD_LEN = [5,3,4,2,5,3,4,2]; record offsets = [0,5,8,12,14,19,22,26]
  dim3 grid(N / NTILE, 2), blk(THREADS);
  group_attn<5><<<grid, blk, 0, stream>>>(x, 0, 14, out, 0, 4, nullptr);
  group_attn<3><<<grid, blk, 0, stream>>>(x, 5, 19, out, 1, 5, nullptr);
  group_attn<4><<<grid, blk, 0, stream>>>(x, 8, 22, out, 2, 6, nullptr);
  group_attn<2><<<grid, blk, 0, stream>>>(x, 12, 26, out, 3, 7, attn);
}